// PPO_16621523435807
// MI455X (gfx1250) — compile-verified
//
#include <hip/hip_runtime.h>
#include <math.h>

// ---------------------------------------------------------------------------
// PPO loss for MI455X (gfx1250, wave32). Memory-bound (~152MB -> ~6.5us floor
// at 23.3 TB/s). All math stays f32; V_WMMA_F32_16X16X4_F32 is used as an
// exact, deterministic f32 cross-lane reduction engine (ones-matrix trick).
// Single-use 128MB streams use non-temporal loads so the re-read data
// (rewards/dones/ret, ~12MB) stays resident in the 192MB L2.
// ---------------------------------------------------------------------------

typedef __attribute__((ext_vector_type(2))) float v2f;
typedef __attribute__((ext_vector_type(4))) float v4f;
typedef __attribute__((ext_vector_type(8))) float v8f;

#define PPO_GAMMA   0.99f
#define PPO_CLIP_LO 0.8f
#define PPO_CLIP_HI 1.2f
#define PPO_LOG2PI  1.8378770664093453f
#define PPO_NC      1024   // scan chunks (== threads in compose kernel)

// Sum of the 64 values {accA,accB} x 32 lanes, broadcast to every lane.
// Pass 1: D = A x ones  -> D[m][*] = row-sum r[m] of the 16x4 accumulator
//         matrix; per C/D layout lane L's 8 VGPRs hold r[0..7] (L<16) or
//         r[8..15] (L>=16); in-lane add gives Rlow / Rhigh.
// Pass 2: A2 = {Rlow|Rhigh, 0} x ones -> every D2 element = Rlow + Rhigh.
__device__ __forceinline__ float wave_sum_wmma(float accA, float accB) {
    v2f a;    a.x = accA; a.y = accB;
    v2f ones; ones.x = 1.0f; ones.y = 1.0f;
    v8f c = {};
    v8f d = __builtin_amdgcn_wmma_f32_16x16x4_f32(
        false, a, false, ones, (short)0, c, false, false);
    float p = d[0] + d[1] + d[2] + d[3] + d[4] + d[5] + d[6] + d[7];
    v2f a2; a2.x = p; a2.y = 0.0f;
    v8f d2 = __builtin_amdgcn_wmma_f32_16x16x4_f32(
        false, a2, false, ones, (short)0, c, false, false);
    return d2[0];
}

// ---- Kernel A: per-chunk affine transfer function (a,b): carry = b + a*x ---
__global__ void ppo_scan_pairs(const float* __restrict__ rewards,
                               const int*   __restrict__ dones,
                               float* __restrict__ ca, float* __restrict__ cb,
                               int L) {
    int c  = blockIdx.x * blockDim.x + threadIdx.x;
    int lo = c * L;
    float a = 1.0f, b = 0.0f;
    for (int t = lo + L - 1; t >= lo; --t) {
        float g = PPO_GAMMA * (1.0f - (float)dones[t]);
        b = fmaf(g, b, rewards[t]);
        a *= g;
    }
    ca[c] = a;
    cb[c] = b;
}

// ---- Kernel B: suffix scan of affine compositions (Hillis-Steele in LDS) ---
__global__ void ppo_scan_compose(const float* __restrict__ ca,
                                 const float* __restrict__ cb,
                                 float* __restrict__ cin, int nc) {
    __shared__ float sa[PPO_NC];
    __shared__ float sb[PPO_NC];
    int tid = threadIdx.x;
    float a = ca[tid], b = cb[tid];
    sa[tid] = a; sb[tid] = b;
    __syncthreads();
    for (int d = 1; d < nc; d <<= 1) {
        float na = a, nb = b;
        if (tid + d < nc) {              // compose with suffix on the right
            na = a * sa[tid + d];
            nb = fmaf(a, sb[tid + d], b);
        }
        __syncthreads();
        sa[tid] = na; sb[tid] = nb;
        a = na; b = nb;
        __syncthreads();
    }
    // carry entering chunk tid from the right = inclusive suffix of tid+1
    cin[tid] = (tid + 1 < nc) ? sb[tid + 1] : 0.0f;
}

// ---- Kernel C: materialize returns + accumulate sum / sumsq ----------------
__global__ void ppo_scan_fill(const float* __restrict__ rewards,
                              const int*   __restrict__ dones,
                              const float* __restrict__ cin,
                              float* __restrict__ ret,
                              float* __restrict__ psum, float* __restrict__ psq,
                              int L) {
    int c  = blockIdx.x * blockDim.x + threadIdx.x;
    int lo = c * L;
    float carry = cin[c];
    float s = 0.0f, ss = 0.0f;
    for (int t = lo + L - 1; t >= lo; --t) {
        float g = PPO_GAMMA * (1.0f - (float)dones[t]);
        carry = fmaf(g, carry, rewards[t]);
        ret[t] = carry;                   // RT store: re-read by ppo_loss (L2 hit)
        s  += carry;
        ss  = fmaf(carry, carry, ss);
    }
    float S  = wave_sum_wmma(s, 0.0f);
    float SS = wave_sum_wmma(ss, 0.0f);
    if ((threadIdx.x & 31) == 0) {
        int w = c >> 5;
        psum[w] = S;
        psq[w]  = SS;
    }
}

// ---- Kernel D: finalize mean / inv(std+eps) (single wave) ------------------
__global__ void ppo_stats(const float* __restrict__ psum,
                          const float* __restrict__ psq,
                          float* __restrict__ stats, float Tf) {
    int lane = threadIdx.x;                 // exactly 32 partials
    float S  = wave_sum_wmma(psum[lane], 0.0f);
    float SS = wave_sum_wmma(psq[lane], 0.0f);
    if (lane == 0) {
        float mean = S / Tf;
        float var  = fmaxf((SS - S * mean) / (Tf - 1.0f), 0.0f);  // ddof=1
        stats[0] = mean;
        stats[1] = 1.0f / (sqrtf(var) + 1e-6f);
    }
}

// ---- per-(t,a) loss term ---------------------------------------------------
__device__ __forceinline__ float ppo_elem(float act, float m, float v,
                                          float olp, float adv) {
    float lv    = logf(v);
    float dd    = act - m;
    float lp    = -0.5f * dd * dd / v - 0.5f * (PPO_LOG2PI + lv);
    float ratio = expf(lp - olp);
    float rc    = fminf(fmaxf(ratio, PPO_CLIP_LO), PPO_CLIP_HI);
    float actor = fminf(ratio * adv, rc * adv);
    // -actor + COEF_ENTROPY * 0.5 * (log2pi + 1 + log var)
    return -actor + 0.005f * (PPO_LOG2PI + 1.0f + lv);
}

// ---- Kernel E: main streaming pass (128MB of [T,8] + [T] arrays) -----------
__global__ void ppo_loss(const float* __restrict__ ret,
                         const float* __restrict__ values_old,
                         const float* __restrict__ values,
                         const float* __restrict__ old_actions,
                         const float* __restrict__ old_log_prob,
                         const float* __restrict__ mu,
                         const float* __restrict__ var,
                         const float* __restrict__ stats,
                         float* __restrict__ lpart, int T) {
    const float mean  = stats[0];
    const float scale = stats[1];
    const v4f* mu4  = (const v4f*)mu;
    const v4f* var4 = (const v4f*)var;
    const v4f* act4 = (const v4f*)old_actions;
    const v4f* olp4 = (const v4f*)old_log_prob;

    int tid    = blockIdx.x * blockDim.x + threadIdx.x;
    int stride = gridDim.x * blockDim.x;
    float acc = 0.0f;

    for (int t = tid; t < T; t += stride) {       // uniform trip count (T%stride==0)
        // ret: RT load (produced by previous kernel, L2 resident)
        float r   = (ret[t] - mean) * scale;      // standardized return
        // single-use [T] streams: non-temporal
        float vo  = __builtin_nontemporal_load(values_old + t);
        float vv  = __builtin_nontemporal_load(values + t);
        float adv = r - vo;
        float dv  = vv - r;
        acc = fmaf(4.0f, dv * dv, acc);           // 8 * (0.5 * critic)

        int b = t * 2;                            // row = 2 x 128-bit loads
        // single-use 32MB streams: non-temporal B128 loads (th:NT)
        v4f m0 = __builtin_nontemporal_load(mu4 + b);
        v4f m1 = __builtin_nontemporal_load(mu4 + b + 1);
        v4f v0 = __builtin_nontemporal_load(var4 + b);
        v4f v1 = __builtin_nontemporal_load(var4 + b + 1);
        v4f a0 = __builtin_nontemporal_load(act4 + b);
        v4f a1 = __builtin_nontemporal_load(act4 + b + 1);
        v4f o0 = __builtin_nontemporal_load(olp4 + b);
        v4f o1 = __builtin_nontemporal_load(olp4 + b + 1);

        acc += ppo_elem(a0.x, m0.x, v0.x, o0.x, adv);
        acc += ppo_elem(a0.y, m0.y, v0.y, o0.y, adv);
        acc += ppo_elem(a0.z, m0.z, v0.z, o0.z, adv);
        acc += ppo_elem(a0.w, m0.w, v0.w, o0.w, adv);
        acc += ppo_elem(a1.x, m1.x, v1.x, o1.x, adv);
        acc += ppo_elem(a1.y, m1.y, v1.y, o1.y, adv);
        acc += ppo_elem(a1.z, m1.z, v1.z, o1.z, adv);
        acc += ppo_elem(a1.w, m1.w, v1.w, o1.w, adv);
    }

    float total = wave_sum_wmma(acc, 0.0f);
    if ((threadIdx.x & 31) == 0)
        lpart[tid >> 5] = total;
}

// ---- Kernel F: final reduction of wave partials (single wave) --------------
__global__ void ppo_final(const float* __restrict__ lpart, float* __restrict__ out,
                          int npart, float inv_n) {
    int lane = threadIdx.x;
    float accA = 0.0f, accB = 0.0f;
    for (int k = 0; k < npart; k += 64) {
        accA += lpart[k + lane];
        accB += lpart[k + 32 + lane];
    }
    float total = wave_sum_wmma(accA, accB);
    if (lane == 0)
        out[0] = total * inv_n;
}

// ---------------------------------------------------------------------------
extern "C" void kernel_launch(void* const* d_in, const int* in_sizes, int n_in,
                              void* d_out, int out_size, void* d_ws, size_t ws_size,
                              hipStream_t stream) {
    const float* rewards      = (const float*)d_in[0];
    const float* values_old   = (const float*)d_in[1];
    const float* old_actions  = (const float*)d_in[2];
    const float* old_log_prob = (const float*)d_in[3];
    const float* mu           = (const float*)d_in[4];
    const float* var          = (const float*)d_in[5];
    const float* values       = (const float*)d_in[6];
    const int*   dones        = (const int*)d_in[7];

    const int T = in_sizes[0];          // 1048576
    const int A = in_sizes[2] / T;      // 8
    const int L = T / PPO_NC;           // 1024 elements per scan chunk

    // workspace layout (floats)
    float* ws    = (float*)d_ws;
    float* ca    = ws;                  // [NC] chunk coef
    float* cb    = ca + PPO_NC;         // [NC] chunk sum
    float* cin   = cb + PPO_NC;         // [NC] chunk incoming carry
    float* ret   = cin + PPO_NC;        // [T]  returns
    float* psum  = ret + T;             // [32] stat partials (sum)
    float* psq   = psum + 32;           // [32] stat partials (sumsq)
    float* stats = psq + 32;            // [2]  mean, 1/(std+eps)
    float* lpart = stats + 8;           // [4096] loss wave partials

    ppo_scan_pairs  <<<PPO_NC / 256, 256, 0, stream>>>(rewards, dones, ca, cb, L);
    ppo_scan_compose<<<1, PPO_NC,        0, stream>>>(ca, cb, cin, PPO_NC);
    ppo_scan_fill   <<<PPO_NC / 256, 256, 0, stream>>>(rewards, dones, cin, ret,
                                                       psum, psq, L);
    ppo_stats       <<<1, 32,            0, stream>>>(psum, psq, stats, (float)T);

    const int EB = 512, ET = 256;                       // 4096 waves
    ppo_loss        <<<EB, ET,           0, stream>>>(ret, values_old, values,
                                                      old_actions, old_log_prob,
                                                      mu, var, stats, lpart, T);
    const int npart = EB * ET / 32;
    ppo_final       <<<1, 32,            0, stream>>>(lpart, (float*)d_out, npart,
                                                      1.0f / ((float)T * (float)A));
}